// GRNGNN_21199958573857
// MI455X (gfx1250) — compile-verified
//
#include <hip/hip_runtime.h>
#include <hip/hip_bf16.h>
#include <stddef.h>

// ---------------------------------------------------------------------------
// GraphSAGE (3-layer, mean aggr) + dot-product decode for MI455X (gfx1250).
// fp32 end-to-end; GEMMs use V_WMMA_F32_16X16X4_F32 (wave32 matrix pipe).
// ---------------------------------------------------------------------------

typedef __attribute__((ext_vector_type(2))) float v2f;
typedef __attribute__((ext_vector_type(8))) float v8f;

#define CH 64   // channel count (in = hidden = out)

// -------------------------------- zero fill --------------------------------
__global__ void sage_zero_f32(float* __restrict__ p, size_t n) {
    size_t i = (size_t)blockIdx.x * blockDim.x + threadIdx.x;
    size_t stride = (size_t)gridDim.x * blockDim.x;
    for (; i < n; i += stride) p[i] = 0.0f;
}

// ------------------------------ degree count -------------------------------
__global__ void sage_degree(const int* __restrict__ dst, float* __restrict__ deg, int E) {
    int e = blockIdx.x * blockDim.x + threadIdx.x;
    if (e >= E) return;
    atomicAdd(&deg[dst[e]], 1.0f);
}

// ------------------------- edge scatter (segment sum) ----------------------
// one thread per (edge, 4-channel group): float4 gather + 4 global f32 atomics
__global__ void sage_scatter(const float* __restrict__ h,
                             const int* __restrict__ src,
                             const int* __restrict__ dst,
                             float* __restrict__ agg, int E) {
    long long tid = (long long)blockIdx.x * blockDim.x + threadIdx.x;
    if (tid >= (long long)E * 16) return;
    int e = (int)(tid >> 4);
    int g = (int)(tid & 15);
    int s = src[e], d = dst[e];
    const float4 v = *(const float4*)(h + (size_t)s * CH + g * 4);
    float* p = agg + (size_t)d * CH + g * 4;
    atomicAdd(p + 0, v.x);
    atomicAdd(p + 1, v.y);
    atomicAdd(p + 2, v.z);
    atomicAdd(p + 3, v.w);
}

// --------------------- fused SAGE linear layer (WMMA f32) ------------------
// out[row, :] = (agg[row,:]/max(deg,1)) @ Wl + hin[row,:] @ Wr + b
//              (+ hin residual) (+ ReLU)
// Block = 256 threads = 8 waves; each wave owns a 16-row tile (block: 128 rows).
// Weights live in LDS with K-pairs interleaved so each lane's B-fragment is a
// single ds_load_b64:  ldsW[(k>>1)*128 + 2*col + (k&1)] = W[k][col]
__global__ __launch_bounds__(256) void sage_gemm_wmma(
    const float* __restrict__ hin,   // [N,64] input features (Wr path + residual)
    const float* __restrict__ agg,   // [N,64] unnormalized neighbor sums
    const float* __restrict__ deg,   // [N]
    const float* __restrict__ Wl,    // [64,64] row-major (k, col)
    const float* __restrict__ Wr,    // [64,64]
    const float* __restrict__ bias,  // [64]
    float* __restrict__ out,         // [N,64]
    int N, int do_relu, int do_residual)
{
    __shared__ float ldsWl[CH * CH];
    __shared__ float ldsWr[CH * CH];
    __shared__ float ldsB[CH];

    const int tid = threadIdx.x;

    // ---- stage weights into LDS (pair-interleaved), coalesced float4 loads
    for (int i = tid; i < (CH * CH) / 4; i += 256) {   // 1024 float4 per matrix
        const int k = i >> 4;              // weight row (K index)
        const int c = (i & 15) << 2;       // starting column
        const float4 a = *(const float4*)(Wl + k * CH + c);
        const float4 b = *(const float4*)(Wr + k * CH + c);
        const int p = (k >> 1) * 128 + (k & 1);
        ldsWl[p + 2 * (c + 0)] = a.x;  ldsWr[p + 2 * (c + 0)] = b.x;
        ldsWl[p + 2 * (c + 1)] = a.y;  ldsWr[p + 2 * (c + 1)] = b.y;
        ldsWl[p + 2 * (c + 2)] = a.z;  ldsWr[p + 2 * (c + 2)] = b.z;
        ldsWl[p + 2 * (c + 3)] = a.w;  ldsWr[p + 2 * (c + 3)] = b.w;
    }
    if (tid < CH) ldsB[tid] = bias[tid];
    __syncthreads();

    const int lane = tid & 31;
    const int wave = tid >> 5;
    const int row0 = blockIdx.x * 128 + wave * 16;   // wave-uniform
    if (row0 + 16 > N) return;                       // full 16-row tiles only (N%16==0)

    // A-fragment geometry (16x4 f32): row = lane&15, k-pair = (lane>>4)*2
    const int m   = lane & 15;
    const int kh  = lane >> 4;      // 0 or 1
    const int k0  = kh * 2;
    const int row = row0 + m;

    const float dscale = 1.0f / fmaxf(deg[row], 1.0f);   // mean-aggr normalization

    // C/D init with bias: every acc element in this lane is column j*16+m
    v8f acc[4];
#pragma unroll
    for (int j = 0; j < 4; ++j) {
        const float bv = ldsB[j * 16 + m];
        acc[j] = (v8f){bv, bv, bv, bv, bv, bv, bv, bv};
    }

    const float* hrow = hin + (size_t)row * CH;
    const float* arow = agg + (size_t)row * CH;

#pragma unroll
    for (int kk = 0; kk < 16; ++kk) {                // K = 64 in steps of 4
        const int kb = kk * 4 + k0;                  // even
        const float2 ha = *(const float2*)(hrow + kb);
        const float2 aa = *(const float2*)(arow + kb);
        v2f aH; aH[0] = ha.x;          aH[1] = ha.y;
        v2f aA; aA[0] = aa.x * dscale; aA[1] = aa.y * dscale;
        const int wbase = (kb >> 1) * 128;           // pair-row base in LDS
#pragma unroll
        for (int j = 0; j < 4; ++j) {                // 4 column tiles of 16
            const int cb = j * 16 + m;
            const v2f bL = *(const v2f*)(ldsWl + wbase + 2 * cb);
            const v2f bR = *(const v2f*)(ldsWr + wbase + 2 * cb);
            acc[j] = __builtin_amdgcn_wmma_f32_16x16x4_f32(
                false, aA, false, bL, (short)0, acc[j], false, false);
            acc[j] = __builtin_amdgcn_wmma_f32_16x16x4_f32(
                false, aH, false, bR, (short)0, acc[j], false, false);
        }
    }

    // ---- epilogue: residual, ReLU, store (coalesced across lanes 0..15)
#pragma unroll
    for (int j = 0; j < 4; ++j) {
        const int col = j * 16 + m;
#pragma unroll
        for (int r = 0; r < 8; ++r) {
            const int mrow = row0 + r + kh * 8;      // C/D layout: M = r + 8*kh
            float v = acc[j][r];
            if (do_residual) v += hin[(size_t)mrow * CH + col];
            if (do_relu)     v = fmaxf(v, 0.0f);
            out[(size_t)mrow * CH + col] = v;
        }
    }
}

// ------------------------------ edge decode --------------------------------
__global__ void sage_decode(const float* __restrict__ z,
                            const int* __restrict__ eli,
                            float* __restrict__ out, int L) {
    int t = blockIdx.x * blockDim.x + threadIdx.x;
    if (t >= L) return;
    const int u = eli[t];
    const int v = eli[L + t];
    const float4* zu = (const float4*)(z + (size_t)u * CH);
    const float4* zv = (const float4*)(z + (size_t)v * CH);
    float s = 0.0f;
#pragma unroll
    for (int i = 0; i < CH / 4; ++i) {
        const float4 a = zu[i], b = zv[i];
        s += a.x * b.x + a.y * b.y + a.z * b.z + a.w * b.w;
    }
    out[t] = s;
}

// ---------------------------------------------------------------------------
extern "C" void kernel_launch(void* const* d_in, const int* in_sizes, int n_in,
                              void* d_out, int out_size, void* d_ws, size_t ws_size,
                              hipStream_t stream) {
    const float* x   = (const float*)d_in[0];
    const int*   ei  = (const int*)d_in[1];
    const int*   eli = (const int*)d_in[2];
    const float* Wl0 = (const float*)d_in[3];
    const float* Wr0 = (const float*)d_in[4];
    const float* b0  = (const float*)d_in[5];
    const float* Wl1 = (const float*)d_in[6];
    const float* Wr1 = (const float*)d_in[7];
    const float* b1  = (const float*)d_in[8];
    const float* Wl2 = (const float*)d_in[9];
    const float* Wr2 = (const float*)d_in[10];
    const float* b2  = (const float*)d_in[11];
    float* out = (float*)d_out;

    const int N = in_sizes[0] / CH;     // 100000
    const int E = in_sizes[1] / 2;      // 1600000
    const int L = in_sizes[2] / 2;      // 200000
    const int* src = ei;
    const int* dst = ei + E;

    // workspace layout (floats): deg[N] | agg[N*64] | hA[N*64] | hB[N*64]
    float* ws  = (float*)d_ws;
    float* deg = ws;
    float* agg = deg + (size_t)N;
    float* hA  = agg + (size_t)N * CH;
    float* hB  = hA  + (size_t)N * CH;

    const size_t nfeat = (size_t)N * CH;
    const int gemm_blocks    = (N + 127) / 128;
    const int scatter_blocks = (int)(((long long)E * 16 + 255) / 256);

    // degree (shared by all layers)
    sage_zero_f32<<<512, 256, 0, stream>>>(deg, (size_t)N);
    sage_degree<<<(E + 255) / 256, 256, 0, stream>>>(dst, deg, E);

    // layer 0: h = relu(mean_agg(x) @ Wl0 + x @ Wr0 + b0)
    sage_zero_f32<<<2048, 256, 0, stream>>>(agg, nfeat);
    sage_scatter<<<scatter_blocks, 256, 0, stream>>>(x, src, dst, agg, E);
    sage_gemm_wmma<<<gemm_blocks, 256, 0, stream>>>(x, agg, deg, Wl0, Wr0, b0,
                                                    hA, N, /*relu=*/1, /*res=*/0);

    // layer 1: h = relu(mean_agg(h) @ Wl1 + h @ Wr1 + b1 + h)
    sage_zero_f32<<<2048, 256, 0, stream>>>(agg, nfeat);
    sage_scatter<<<scatter_blocks, 256, 0, stream>>>(hA, src, dst, agg, E);
    sage_gemm_wmma<<<gemm_blocks, 256, 0, stream>>>(hA, agg, deg, Wl1, Wr1, b1,
                                                    hB, N, /*relu=*/1, /*res=*/1);

    // layer 2: z = mean_agg(h) @ Wl2 + h @ Wr2 + b2
    sage_zero_f32<<<2048, 256, 0, stream>>>(agg, nfeat);
    sage_scatter<<<scatter_blocks, 256, 0, stream>>>(hB, src, dst, agg, E);
    sage_gemm_wmma<<<gemm_blocks, 256, 0, stream>>>(hB, agg, deg, Wl2, Wr2, b2,
                                                    hA, N, /*relu=*/0, /*res=*/0);

    // decode: out[t] = dot(z[u_t], z[v_t])
    sage_decode<<<(L + 255) / 256, 256, 0, stream>>>(hA, eli, out, L);
}